// BiLRNN_4063039062786
// MI455X (gfx1250) — compile-verified
//
#include <hip/hip_runtime.h>
#include <math.h>

#define N_TOK 8192
#define DF    256
#define CSZ   32
#define KSZ   128
#define NCH   16
#define HALF  64

typedef __attribute__((ext_vector_type(2))) float v2f;
typedef __attribute__((ext_vector_type(8))) float v8f;

// ---------------------------------------------------------------------------
// Phase 0: chain start offsets from sorted index. Each boundary writes a
// distinct set of start[] slots -> race-free, deterministic.
// ---------------------------------------------------------------------------
__global__ void starts_kernel(const int* __restrict__ idx, int* __restrict__ start) {
    int n = blockIdx.x * blockDim.x + threadIdx.x;
    if (n >= N_TOK) return;
    int v  = idx[n];
    int vp = (n == 0) ? -1 : idx[n - 1];
    for (int w = vp + 1; w <= v; ++w) start[w] = n;
    if (n == N_TOK - 1)
        for (int w = v + 1; w <= NCH; ++w) start[w] = N_TOK;
}

// ---------------------------------------------------------------------------
// Phase 1: all five projections via V_WMMA_F32_16X16X4_F32.
// grid = (N/16 row tiles, 28 col tiles), block = 1 wave (32 lanes).
// Col tiles: [0,8)=W_gate->gs(+nonlinearity) [8,16)=W_gproj->gp
//            [16,18)=W_gchan->gc [18,20)=W_valc->vc [20,28)=W_vals->vs
// ---------------------------------------------------------------------------
__global__ __launch_bounds__(32) void proj_kernel(
    const float* __restrict__ data,
    const float* __restrict__ Wgate,  const float* __restrict__ Wgproj,
    const float* __restrict__ Wgchan, const float* __restrict__ Wvalc,
    const float* __restrict__ Wvals,  const float* __restrict__ decay_log,
    float* __restrict__ gs, float* __restrict__ gp, float* __restrict__ gc,
    float* __restrict__ vc, float* __restrict__ vs)
{
    const int rt   = blockIdx.x;        // row tile
    const int ct   = blockIdx.y;        // col tile (0..27)
    const int lane = threadIdx.x;       // 0..31
    const int m    = lane & 15;         // A row / D col within tile
    const int hi   = lane >> 4;

    const float* W; float* Out; int wcols, col0, is_gate;
    if      (ct <  8) { W = Wgate;  Out = gs; wcols = KSZ; col0 = ct*16;       is_gate = 1; }
    else if (ct < 16) { W = Wgproj; Out = gp; wcols = KSZ; col0 = (ct-8)*16;   is_gate = 0; }
    else if (ct < 18) { W = Wgchan; Out = gc; wcols = CSZ; col0 = (ct-16)*16;  is_gate = 0; }
    else if (ct < 20) { W = Wvalc;  Out = vc; wcols = CSZ; col0 = (ct-18)*16;  is_gate = 0; }
    else              { W = Wvals;  Out = vs; wcols = KSZ; col0 = (ct-20)*16;  is_gate = 0; }

    const float* arow = data + (size_t)(rt*16 + m) * DF;   // A: M = lane&15
    const float* bcol = W + col0 + m;                      // B: N = lane&15

    v8f acc = {0.f,0.f,0.f,0.f,0.f,0.f,0.f,0.f};
    #pragma unroll 4
    for (int k0 = 0; k0 < DF; k0 += 4) {
        const int ka = k0 + 2*hi;                          // K pair per lane half
        v2f a; a.x = arow[ka];                    a.y = arow[ka + 1];
        v2f b; b.x = bcol[(size_t)ka * wcols];    b.y = bcol[(size_t)(ka+1) * wcols];
        acc = __builtin_amdgcn_wmma_f32_16x16x4_f32(
                  /*neg_a=*/false, a, /*neg_b=*/false, b,
                  /*c_mod=*/(short)0, acc, /*reuse_a=*/false, /*reuse_b=*/false);
    }

    // D layout: lane n = lane&15, rows r + 8*hi in acc[r]
    const int col = col0 + m;
    float db = 0.f;
    if (is_gate) db = __expf(decay_log[col]);
    #pragma unroll
    for (int r = 0; r < 8; ++r) {
        const int row = rt*16 + r + 8*hi;
        float v = acc[r];
        if (is_gate) v = __expf(-__expf(v) - db);          // gate_s
        Out[(size_t)row * wcols + col] = v;
    }
}

// ---------------------------------------------------------------------------
// Phase 2: bidirectional per-chain scan, fused rank-1 update + contraction.
// grid = (16 chains, 2 dirs), block = 256.  Thread t owns c = t>>3 and the
// 8 k-values k = (t&7)*8 .. +7 (within the direction's half).  State lives
// in registers; per-step k-reduction via wave32 shfl_xor over 8 lanes.
// Backward dir: gates/values from token N-1-p, gp/out at scan position p.
// ---------------------------------------------------------------------------
__global__ __launch_bounds__(256) void scan_kernel(
    const float* __restrict__ gs, const float* __restrict__ gp,
    const float* __restrict__ gc, const float* __restrict__ vc,
    const float* __restrict__ vs, const int* __restrict__ start,
    float* __restrict__ outF, float* __restrict__ outB)
{
    const int ch  = blockIdx.x;
    const int dir = blockIdx.y;
    const int s = start[ch], e = start[ch + 1];
    const int len = e - s;
    const int tid   = threadIdx.x;
    const int c     = tid >> 3;
    const int kg    = tid & 7;
    const int kglob = dir * HALF + kg * 8;

    float S[8];
    #pragma unroll
    for (int j = 0; j < 8; ++j) S[j] = 0.f;

    float* __restrict__ out = dir ? outB : outF;

    for (int i = 0; i < len; ++i) {
        const int t = dir ? (e - 1 - i) : (s + i);     // token for gates/values
        const int p = dir ? (N_TOK - 1 - t) : t;       // gp / output position

        const float4* gsr = (const float4*)(gs + (size_t)t * KSZ + kglob);
        const float4* vsr = (const float4*)(vs + (size_t)t * KSZ + kglob);
        const float4* gpr = (const float4*)(gp + (size_t)p * KSZ + kglob);
        float4 g0 = gsr[0], g1 = gsr[1];
        float4 v0 = vsr[0], v1 = vsr[1];
        float4 q0 = gpr[0], q1 = gpr[1];
        const float gcv = gc[(size_t)t * CSZ + c];
        const float vcv = vc[(size_t)t * CSZ + c];

        if (i + 1 < len) {                             // prefetch next token rows
            const int tn = dir ? (t - 1) : (t + 1);
            __builtin_prefetch(gs + (size_t)tn * KSZ + kglob, 0, 0);
            __builtin_prefetch(vs + (size_t)tn * KSZ + kglob, 0, 0);
        }

        const float gsv[8] = {g0.x,g0.y,g0.z,g0.w,g1.x,g1.y,g1.z,g1.w};
        const float vsv[8] = {v0.x,v0.y,v0.z,v0.w,v1.x,v1.y,v1.z,v1.w};
        const float gpv[8] = {q0.x,q0.y,q0.z,q0.w,q1.x,q1.y,q1.z,q1.w};

        float partial = 0.f;
        #pragma unroll
        for (int j = 0; j < 8; ++j) {
            S[j] = gsv[j] * gcv * S[j] + vcv * vsv[j];
            partial = __builtin_fmaf(S[j], gpv[j], partial);
        }
        // reduce across the 8 lanes sharing c (low 3 lane bits)
        partial += __shfl_xor(partial, 1, 32);
        partial += __shfl_xor(partial, 2, 32);
        partial += __shfl_xor(partial, 4, 32);
        if (kg == 0) out[(size_t)p * CSZ + c] = partial;
    }
}

// ---------------------------------------------------------------------------
// Phase 3: out = outF + outB
// ---------------------------------------------------------------------------
__global__ void add_kernel(const float* __restrict__ a, const float* __restrict__ b,
                           float* __restrict__ o, int n) {
    int i = blockIdx.x * blockDim.x + threadIdx.x;
    if (i < n) o[i] = a[i] + b[i];
}

// ---------------------------------------------------------------------------
extern "C" void kernel_launch(void* const* d_in, const int* in_sizes, int n_in,
                              void* d_out, int out_size, void* d_ws, size_t ws_size,
                              hipStream_t stream) {
    const float* data      = (const float*)d_in[0];
    const int*   index     = (const int*)  d_in[1];
    const float* decay_log = (const float*)d_in[2];
    const float* Wgate     = (const float*)d_in[3];
    const float* Wgproj    = (const float*)d_in[4];
    const float* Wgchan    = (const float*)d_in[5];
    const float* Wvalc     = (const float*)d_in[6];
    const float* Wvals     = (const float*)d_in[7];

    float* ws   = (float*)d_ws;
    float* gs   = ws;                       // N x 128
    float* gp   = gs   + (size_t)N_TOK*KSZ; // N x 128
    float* vs   = gp   + (size_t)N_TOK*KSZ; // N x 128
    float* gc   = vs   + (size_t)N_TOK*KSZ; // N x 32
    float* vc   = gc   + (size_t)N_TOK*CSZ; // N x 32
    float* outF = vc   + (size_t)N_TOK*CSZ; // N x 32
    float* outB = outF + (size_t)N_TOK*CSZ; // N x 32
    int*   start = (int*)(outB + (size_t)N_TOK*CSZ); // 17 ints

    starts_kernel<<<N_TOK / 256, 256, 0, stream>>>(index, start);

    proj_kernel<<<dim3(N_TOK / 16, 28), 32, 0, stream>>>(
        data, Wgate, Wgproj, Wgchan, Wvalc, Wvals, decay_log,
        gs, gp, gc, vc, vs);

    scan_kernel<<<dim3(NCH, 2), 256, 0, stream>>>(
        gs, gp, gc, vc, vs, start, outF, outB);

    add_kernel<<<(N_TOK * CSZ + 255) / 256, 256, 0, stream>>>(
        outF, outB, (float*)d_out, N_TOK * CSZ);
}